// GATv4_43782896615769
// MI455X (gfx1250) — compile-verified
//
#include <hip/hip_runtime.h>

// ---------------- problem constants (match reference) ----------------
#define NTOT   32000      // total nodes (B*NN)
#define NNODE  1000       // nodes per graph
#define BATCH  32
#define NEDGE  512000     // edges before self loops
#define ENTOT  (NEDGE + NTOT)
#define NHEAD  4
#define NCH    64
#define NHC    256        // NHEAD*NCH
#define NIN    128
#define NFC    512
#define LN_EPS 1e-5f

typedef __attribute__((ext_vector_type(2))) float v2f;
typedef __attribute__((ext_vector_type(8))) float v8f;

// ---------------- fp32 WMMA GEMM:  Out[N,256] = X[N,K] @ W[256,K]^T ----------------
// One wave per 16x64 output tile. grid = (N/16, 4), block = 32 (wave32, EXEC all ones).
// A-fragment (16x4 f32): lanes 0-15 -> M=lane, K={k0,k0+1}; lanes 16-31 -> M=lane-16, K={k0+2,k0+3}
// B-fragment (4x16 f32): lanes 0-15 -> N=lane, K={k0,k0+1}; lanes 16-31 -> N=lane-16, K={k0+2,k0+3}
template<int K>
__global__ __launch_bounds__(32) void gemm_wmma(const float* __restrict__ X,
                                                const float* __restrict__ W,
                                                float* __restrict__ Out) {
  const int lane = threadIdx.x;
  const int mt   = blockIdx.x;          // M tile (16 rows)
  const int cg   = blockIdx.y;          // 64-column group
  const int r    = lane & 15;
  const int kh   = lane >> 4;           // 0 or 1
  const float* xrow = X + (size_t)(mt * 16 + r) * K;
  const int colb = cg * 64 + r;
  v8f acc0 = {}; v8f acc1 = {}; v8f acc2 = {}; v8f acc3 = {};
  for (int k0 = 0; k0 < K; k0 += 4) {
    const int ko = k0 + 2 * kh;
    v2f a  = *(const v2f*)(xrow + ko);
    v2f b0 = *(const v2f*)(W + (size_t)(colb +  0) * K + ko);
    v2f b1 = *(const v2f*)(W + (size_t)(colb + 16) * K + ko);
    v2f b2 = *(const v2f*)(W + (size_t)(colb + 32) * K + ko);
    v2f b3 = *(const v2f*)(W + (size_t)(colb + 48) * K + ko);
    acc0 = __builtin_amdgcn_wmma_f32_16x16x4_f32(false, a, false, b0, (short)0, acc0, false, false);
    acc1 = __builtin_amdgcn_wmma_f32_16x16x4_f32(false, a, false, b1, (short)0, acc1, false, false);
    acc2 = __builtin_amdgcn_wmma_f32_16x16x4_f32(false, a, false, b2, (short)0, acc2, false, false);
    acc3 = __builtin_amdgcn_wmma_f32_16x16x4_f32(false, a, false, b3, (short)0, acc3, false, false);
  }
  // C/D layout: VGPR i -> M = 8*(lane>=16) + i, N = lane&15
  const int mb = (lane >> 4) * 8;
  float* obase = Out + (size_t)(mt * 16 + mb) * NHC + cg * 64 + (lane & 15);
  #pragma unroll
  for (int i = 0; i < 8; ++i) {
    float* o = obase + (size_t)i * NHC;
    o[0]  = acc0[i];
    o[16] = acc1[i];
    o[32] = acc2[i];
    o[48] = acc3[i];
  }
}

// ---------------- per-node attention projections: a_src/a_dst [N,4] ----------------
__global__ void attn_proj(const float* __restrict__ Hf, const float* __restrict__ attS,
                          const float* __restrict__ attD, float* __restrict__ asrc,
                          float* __restrict__ adst) {
  int idx = blockIdx.x * blockDim.x + threadIdx.x;
  if (idx >= NTOT * NHEAD) return;
  int n = idx >> 2, h = idx & 3;
  const float* hp = Hf + (size_t)n * NHC + h * NCH;
  const float* ws = attS + h * NCH;
  const float* wd = attD + h * NCH;
  float s = 0.f, d = 0.f;
  #pragma unroll 8
  for (int c = 0; c < NCH; ++c) { s += hp[c] * ws[c]; d += hp[c] * wd[c]; }
  asrc[idx] = s; adst[idx] = d;
}

// ---------------- init per-layer stats & aggregation buffer ----------------
__global__ void init_layer(float* __restrict__ m, float* __restrict__ s, float* __restrict__ agg) {
  int i = blockIdx.x * blockDim.x + threadIdx.x;
  if (i < NTOT * NHEAD) { m[i] = -3.0e38f; s[i] = 0.f; }
  if (i < NTOT * NHC) agg[i] = 0.f;
}

__device__ inline float leaky02(float x) { return x > 0.f ? x : 0.2f * x; }

__device__ inline void atomicMaxF(float* addr, float val) {
  int iv = __float_as_int(val);
  if (val >= 0.f) atomicMax((int*)addr, iv);
  else            atomicMin((unsigned int*)addr, (unsigned int)iv);
}

__device__ inline void edge_sd(int e, const int* __restrict__ src, const int* __restrict__ dst,
                               int& s, int& d) {
  if (e < NEDGE) { s = src[e]; d = dst[e]; }
  else           { s = e - NEDGE; d = s; }          // implicit self-loops
}

// ---------------- pass 1: segment max of leaky-relu logits ----------------
__global__ void edge_max(const int* __restrict__ src, const int* __restrict__ dst,
                         const float* __restrict__ asrc, const float* __restrict__ adst,
                         float* __restrict__ mmax) {
  int idx = blockIdx.x * blockDim.x + threadIdx.x;     // over ENTOT*NHEAD
  if (idx >= ENTOT * NHEAD) return;
  int e = idx >> 2, h = idx & 3, s, d;
  edge_sd(e, src, dst, s, d);
  float l = leaky02(asrc[s * NHEAD + h] + adst[d * NHEAD + h]);
  atomicMaxF(&mmax[d * NHEAD + h], l);
}

// ---------------- pass 2: segment sum of exp(logit - max) ----------------
__global__ void edge_sum(const int* __restrict__ src, const int* __restrict__ dst,
                         const float* __restrict__ asrc, const float* __restrict__ adst,
                         const float* __restrict__ mmax, float* __restrict__ ssum) {
  int idx = blockIdx.x * blockDim.x + threadIdx.x;
  if (idx >= ENTOT * NHEAD) return;
  int e = idx >> 2, h = idx & 3, s, d;
  edge_sd(e, src, dst, s, d);
  float l = leaky02(asrc[s * NHEAD + h] + adst[d * NHEAD + h]);
  atomicAdd(&ssum[d * NHEAD + h], __expf(l - mmax[d * NHEAD + h]));
}

// ---------------- pass 3: agg[dst] += h[src] * alpha  (wave per edge) ----------------
__global__ void edge_agg(const int* __restrict__ src, const int* __restrict__ dst,
                         const float* __restrict__ asrc, const float* __restrict__ adst,
                         const float* __restrict__ mmax, const float* __restrict__ ssum,
                         const float* __restrict__ Hf, float* __restrict__ agg) {
  int e = blockIdx.x * (blockDim.x >> 5) + (threadIdx.x >> 5);
  if (e >= ENTOT) return;
  int lane = threadIdx.x & 31;
  int s, d;
  edge_sd(e, src, dst, s, d);
  int h = lane >> 3;                                   // 8 lanes per head
  float l = leaky02(asrc[s * NHEAD + h] + adst[d * NHEAD + h]);
  float alpha = __expf(l - mmax[d * NHEAD + h]) / ssum[d * NHEAD + h];
  int ch = lane * 8;
  const float* hs = Hf + (size_t)s * NHC + ch;
  float* od = agg + (size_t)d * NHC + ch;
  #pragma unroll
  for (int i = 0; i < 8; ++i) atomicAdd(od + i, hs[i] * alpha);
}

// ---------------- epilogue: bias + ELU, store hact, pool to scalar ----------------
__global__ __launch_bounds__(256) void epilogue(const float* __restrict__ agg,
                                                const float* __restrict__ bias,
                                                const float* __restrict__ pW,
                                                const float* __restrict__ pb,
                                                float* __restrict__ hact,
                                                float* __restrict__ xpool) {
  __shared__ float red[8];
  int n = blockIdx.x, t = threadIdx.x;
  float v = agg[(size_t)n * NHC + t] + bias[t];
  v = v > 0.f ? v : (__expf(v) - 1.f);                 // ELU
  hact[(size_t)n * NHC + t] = v;
  float p = v * pW[t];
  #pragma unroll
  for (int o = 16; o > 0; o >>= 1) p += __shfl_down(p, o, 32);
  if ((t & 31) == 0) red[t >> 5] = p;
  __syncthreads();
  if (t == 0) {
    float q = 0.f;
    #pragma unroll
    for (int i = 0; i < 8; ++i) q += red[i];
    xpool[n] = q + pb[0];
  }
}

// ---------------- x0: per-node mean over 128 input features ----------------
__global__ void mean_x0(const float* __restrict__ x, float* __restrict__ x0) {
  int n = blockIdx.x * blockDim.x + threadIdx.x;
  if (n >= NTOT) return;
  const float* r = x + (size_t)n * NIN;
  float s = 0.f;
  #pragma unroll 8
  for (int i = 0; i < NIN; ++i) s += r[i];
  x0[n] = s * (1.f / NIN);
}

__device__ inline float block_sum256(float v, float* red) {
  #pragma unroll
  for (int o = 16; o > 0; o >>= 1) v += __shfl_down(v, o, 32);
  int t = threadIdx.x;
  if ((t & 31) == 0) red[t >> 5] = v;
  __syncthreads();
  if (t == 0) {
    float q = 0.f;
    #pragma unroll
    for (int i = 0; i < 8; ++i) q += red[i];
    red[0] = q;
  }
  __syncthreads();
  float total = red[0];
  __syncthreads();
  return total;
}

// ---------------- LayerNorm over NN per graph -> feats[B, 3000] slice ----------------
__global__ __launch_bounds__(256) void layernorm_k(const float* __restrict__ v,
                                                   const float* __restrict__ lnw,
                                                   const float* __restrict__ lnb,
                                                   float* __restrict__ feats, int which) {
  __shared__ float red[8];
  int g = blockIdx.x;
  const float* row = v + (size_t)g * NNODE;
  float s = 0.f;
  for (int i = threadIdx.x; i < NNODE; i += 256) s += row[i];
  float mu = block_sum256(s, red) * (1.f / NNODE);
  float s2 = 0.f;
  for (int i = threadIdx.x; i < NNODE; i += 256) { float d = row[i] - mu; s2 += d * d; }
  float var = block_sum256(s2, red) * (1.f / NNODE);
  float inv = rsqrtf(var + LN_EPS);
  float* out = feats + (size_t)g * (3 * NNODE) + which * NNODE;
  for (int i = threadIdx.x; i < NNODE; i += 256)
    out[i] = (row[i] - mu) * inv * lnw[i] + lnb[i];
}

// ---------------- small dense layers ----------------
__global__ void fc_tanh(const float* __restrict__ in, const float* __restrict__ Wm,
                        const float* __restrict__ bias, float* __restrict__ out,
                        int Kd, int Fd) {
  int f = blockIdx.x * blockDim.x + threadIdx.x;
  int b = blockIdx.y;
  if (f >= Fd) return;
  const float* r = in + (size_t)b * Kd;
  const float* w = Wm + (size_t)f * Kd;
  float s = 0.f;
  for (int k = 0; k < Kd; ++k) s += r[k] * w[k];
  out[(size_t)b * Fd + f] = tanhf(s + bias[f]);
}

__global__ void fc_final(const float* __restrict__ z, const float* __restrict__ w,
                         const float* __restrict__ b, float* __restrict__ out) {
  int bi = threadIdx.x;                    // 32 graphs
  const float* r = z + (size_t)bi * NFC;
  float s = 0.f;
  for (int k = 0; k < NFC; ++k) s += r[k] * w[k];
  out[bi] = s + b[0];
}

// ---------------- host orchestration ----------------
extern "C" void kernel_launch(void* const* d_in, const int* in_sizes, int n_in,
                              void* d_out, int out_size, void* d_ws, size_t ws_size,
                              hipStream_t stream) {
  const float* x     = (const float*)d_in[0];
  const int*   ei    = (const int*)  d_in[1];
  const float* W1    = (const float*)d_in[2];
  const float* attS1 = (const float*)d_in[3];
  const float* attD1 = (const float*)d_in[4];
  const float* b1    = (const float*)d_in[5];
  const float* W2    = (const float*)d_in[6];
  const float* attS2 = (const float*)d_in[7];
  const float* attD2 = (const float*)d_in[8];
  const float* b2    = (const float*)d_in[9];
  const float* pW1   = (const float*)d_in[10];
  const float* pb1   = (const float*)d_in[11];
  const float* pW2   = (const float*)d_in[12];
  const float* pb2   = (const float*)d_in[13];
  const float* lnw   = (const float*)d_in[14];
  const float* lnb   = (const float*)d_in[15];
  const float* fcW1  = (const float*)d_in[16];
  const float* fcb1  = (const float*)d_in[17];
  const float* fcW2  = (const float*)d_in[18];
  const float* fcb2  = (const float*)d_in[19];
  const float* fcW3  = (const float*)d_in[20];
  const float* fcb3  = (const float*)d_in[21];
  float* out = (float*)d_out;

  const int* src = ei;
  const int* dst = ei + NEDGE;

  // workspace carve-out (floats)
  float* ws    = (float*)d_ws;
  float* hbuf  = ws;                         // [N,256] GEMM output
  float* agg   = hbuf + (size_t)NTOT * NHC;  // [N,256] aggregation
  float* hact  = agg  + (size_t)NTOT * NHC;  // [N,256] ELU output (layer-2 input)
  float* asrc  = hact + (size_t)NTOT * NHC;  // [N,4]
  float* adst  = asrc + NTOT * NHEAD;
  float* mmax  = adst + NTOT * NHEAD;
  float* ssum  = mmax + NTOT * NHEAD;
  float* x0v   = ssum + NTOT * NHEAD;        // [N]
  float* x1v   = x0v + NTOT;
  float* x2v   = x1v + NTOT;
  float* feats = x2v + NTOT;                 // [32,3000]
  float* z1    = feats + BATCH * 3 * NNODE;  // [32,512]
  float* z2    = z1 + BATCH * NFC;

  const dim3 gemmGrid(NTOT / 16, 4);
  const int  edgeThreads = ENTOT * NHEAD;
  const int  edgeBlocks  = (edgeThreads + 255) / 256;
  const int  aggBlocks   = (ENTOT + 7) / 8;       // 8 waves (edges) per 256-thread block
  const int  initBlocks  = (NTOT * NHC + 255) / 256;

  // ---------- x0 ----------
  mean_x0<<<(NTOT + 255) / 256, 256, 0, stream>>>(x, x0v);

  // ---------- GAT layer 1 ----------
  gemm_wmma<NIN><<<gemmGrid, 32, 0, stream>>>(x, W1, hbuf);
  attn_proj<<<(NTOT * NHEAD + 255) / 256, 256, 0, stream>>>(hbuf, attS1, attD1, asrc, adst);
  init_layer<<<initBlocks, 256, 0, stream>>>(mmax, ssum, agg);
  edge_max<<<edgeBlocks, 256, 0, stream>>>(src, dst, asrc, adst, mmax);
  edge_sum<<<edgeBlocks, 256, 0, stream>>>(src, dst, asrc, adst, mmax, ssum);
  edge_agg<<<aggBlocks, 256, 0, stream>>>(src, dst, asrc, adst, mmax, ssum, hbuf, agg);
  epilogue<<<NTOT, 256, 0, stream>>>(agg, b1, pW1, pb1, hact, x1v);

  // ---------- GAT layer 2 ----------
  gemm_wmma<NHC><<<gemmGrid, 32, 0, stream>>>(hact, W2, hbuf);
  attn_proj<<<(NTOT * NHEAD + 255) / 256, 256, 0, stream>>>(hbuf, attS2, attD2, asrc, adst);
  init_layer<<<initBlocks, 256, 0, stream>>>(mmax, ssum, agg);
  edge_max<<<edgeBlocks, 256, 0, stream>>>(src, dst, asrc, adst, mmax);
  edge_sum<<<edgeBlocks, 256, 0, stream>>>(src, dst, asrc, adst, mmax, ssum);
  edge_agg<<<aggBlocks, 256, 0, stream>>>(src, dst, asrc, adst, mmax, ssum, hbuf, agg);
  epilogue<<<NTOT, 256, 0, stream>>>(agg, b2, pW2, pb2, hact, x2v);

  // ---------- LayerNorm features ----------
  layernorm_k<<<BATCH, 256, 0, stream>>>(x0v, lnw, lnb, feats, 0);
  layernorm_k<<<BATCH, 256, 0, stream>>>(x1v, lnw, lnb, feats, 1);
  layernorm_k<<<BATCH, 256, 0, stream>>>(x2v, lnw, lnb, feats, 2);

  // ---------- FC head ----------
  fc_tanh<<<dim3((NFC + 255) / 256, BATCH), 256, 0, stream>>>(feats, fcW1, fcb1, z1, 3 * NNODE, NFC);
  fc_tanh<<<dim3((NFC + 255) / 256, BATCH), 256, 0, stream>>>(z1, fcW2, fcb2, z2, NFC, NFC);
  fc_final<<<1, 32, 0, stream>>>(z2, fcW3, fcb3, out);
}